// GNNLayers_21105469293027
// MI455X (gfx1250) — compile-verified
//
#include <hip/hip_runtime.h>
#include <cstdint>

typedef __attribute__((ext_vector_type(2))) float v2f;
typedef __attribute__((ext_vector_type(8))) float v8f;

#define RESHAPE 224

// ---------------- utility kernels ----------------

__global__ void zero_f32_kernel(float* __restrict__ p, long long n) {
  long long i = (long long)blockIdx.x * blockDim.x + threadIdx.x;
  if (i < n) p[i] = 0.0f;
}

__global__ void count_deg_kernel(const int* __restrict__ dst, float* __restrict__ deg, int E) {
  int i = blockIdx.x * blockDim.x + threadIdx.x;
  if (i < E) atomicAdd(&deg[dst[i]], 1.0f);
}

// deg includes +1 self-loop; deg >= 1 always so rsqrt is safe.
__global__ void deg_to_dinv_kernel(float* __restrict__ deg, int n) {
  int i = blockIdx.x * blockDim.x + threadIdx.x;
  if (i < n) deg[i] = rsqrtf(deg[i] + 1.0f);
}

// ---------------- WMMA f32 GEMM kernels ----------------
// One wave per 16x16 output tile. M%16==0, N%16==0, K%4==0.
// f32 16x16x4 VGPR layout (ISA 7.12.2):
//   lane L: h = L>>4, c = L&15
//   A vgpr v  -> A[c][2h+v]      (A is 16x4 tile, row = c)
//   B vgpr v  -> B[2h+v][c]      (B is 4x16 tile, col = c)
//   C/D vgpr v -> C[v+8h][c]

// C[m][n] = sum_k A[m][k] * B[k][n];  A: MxK row-major, B: KxN row-major
__global__ void gemm_nn_wmma(const float* __restrict__ A, const float* __restrict__ B,
                             float* __restrict__ C, int M, int N, int K) {
  const int tn = blockIdx.x;          // N/16
  const int tm = blockIdx.y;          // M/16
  const int lane = threadIdx.x;       // 0..31, full wave
  const int h = lane >> 4;
  const int c = lane & 15;
  const int m = tm * 16 + c;          // A row for this lane
  const int n = tn * 16 + c;          // B/C col for this lane

  v8f acc = {};
  const float* arow = A + (size_t)m * K + 2 * h;
  for (int k0 = 0; k0 < K; k0 += 4) {
    v2f a = *(const v2f*)(arow + k0);                 // A[m][k0+2h], A[m][k0+2h+1]
    v2f b;
    b.x = B[(size_t)(k0 + 2 * h) * N + n];            // B[k0+2h][n]
    b.y = B[(size_t)(k0 + 2 * h + 1) * N + n];        // B[k0+2h+1][n]
    acc = __builtin_amdgcn_wmma_f32_16x16x4_f32(false, a, false, b,
                                                (short)0, acc, false, false);
  }
#pragma unroll
  for (int v = 0; v < 8; ++v) {
    C[(size_t)(tm * 16 + v + 8 * h) * N + n] = acc[v];
  }
}

// C[m][n] = sum_k A[m][k] * Bt[n][k];  A: MxK row-major, Bt: NxK row-major
__global__ void gemm_nt_wmma(const float* __restrict__ A, const float* __restrict__ Bt,
                             float* __restrict__ C, int M, int N, int K) {
  const int tn = blockIdx.x;
  const int tm = blockIdx.y;
  const int lane = threadIdx.x;
  const int h = lane >> 4;
  const int c = lane & 15;
  const int m = tm * 16 + c;
  const int n = tn * 16 + c;

  v8f acc = {};
  const float* arow = A + (size_t)m * K + 2 * h;
  const float* brow = Bt + (size_t)n * K + 2 * h;
  for (int k0 = 0; k0 < K; k0 += 4) {
    v2f a = *(const v2f*)(arow + k0);
    v2f b = *(const v2f*)(brow + k0);                 // B[k][n] = Bt[n][k], contiguous pair
    acc = __builtin_amdgcn_wmma_f32_16x16x4_f32(false, a, false, b,
                                                (short)0, acc, false, false);
  }
#pragma unroll
  for (int v = 0; v < 8; ++v) {
    C[(size_t)(tm * 16 + v + 8 * h) * N + n] = acc[v];
  }
}

// ---------------- GCN aggregation ----------------

// agg[dst] += h[src] * dinv[src] * dinv[dst], edge-parallel, D consecutive lanes per edge
__global__ void scatter_edge_kernel(const float* __restrict__ h, const int* __restrict__ src,
                                    const int* __restrict__ dst, const float* __restrict__ dinv,
                                    float* __restrict__ agg, int E, int D) {
  long long idx = (long long)blockIdx.x * blockDim.x + threadIdx.x;
  int e = (int)(idx / D);
  int f = (int)(idx % D);
  if (e >= E) return;
  int s = src[e];
  int t = dst[e];
  float norm = dinv[s] * dinv[t];
  atomicAdd(agg + (size_t)t * D + f, h[(size_t)s * D + f] * norm);
}

// y = relu(agg + h*dinv^2 (self-loop) + bias)
__global__ void finalize_kernel(const float* __restrict__ agg, const float* __restrict__ h,
                                const float* __restrict__ dinv, const float* __restrict__ bias,
                                float* __restrict__ y, int N, int D) {
  long long idx = (long long)blockIdx.x * blockDim.x + threadIdx.x;
  int v = (int)(idx / D);
  int f = (int)(idx % D);
  if (v >= N) return;
  float di = dinv[v];
  float val = agg[idx] + h[idx] * di * di + bias[f];
  y[idx] = fmaxf(val, 0.0f);
}

// ---------------- bilinear row interpolation (half-pixel, align_corners=False) ----------------
// U[i] = (1-fr)*X[r0] + fr*X[r1] with pos = (i+0.5)*in_size/224 - 0.5; rows >= Nvalid are zero pad.
__global__ void interp_rows_kernel(const float* __restrict__ X, int Nvalid, int D,
                                   int in_size, float* __restrict__ U) {
  int i = blockIdx.x;         // 0..223
  int f = threadIdx.x;        // 0..D-1
  if (f >= D) return;
  float pos = (i + 0.5f) * ((float)in_size / (float)RESHAPE) - 0.5f;
  float rf = floorf(pos);
  int r0 = (int)rf;
  float fr = pos - rf;
  int r1 = r0 + 1;
  r0 = r0 < 0 ? 0 : (r0 > in_size - 1 ? in_size - 1 : r0);
  r1 = r1 < 0 ? 0 : (r1 > in_size - 1 ? in_size - 1 : r1);
  float v0 = (r0 < Nvalid) ? X[(size_t)r0 * D + f] : 0.0f;
  float v1 = (r1 < Nvalid) ? X[(size_t)r1 * D + f] : 0.0f;
  U[(size_t)i * D + f] = (1.0f - fr) * v0 + fr * v1;
}

// ---------------- host-side orchestration ----------------

static void run_gcn_layer(const float* xin, int N, int K, int D,
                          const int* src, const int* dst, int E,
                          const float* W, const float* b, const float* dinv,
                          float* h, float* agg, float* y, hipStream_t stream) {
  dim3 gg(D / 16, N / 16);
  gemm_nn_wmma<<<gg, 32, 0, stream>>>(xin, W, h, N, D, K);
  long long nAgg = (long long)N * D;
  zero_f32_kernel<<<(unsigned)((nAgg + 255) / 256), 256, 0, stream>>>(agg, nAgg);
  long long nEd = (long long)E * D;
  scatter_edge_kernel<<<(unsigned)((nEd + 255) / 256), 256, 0, stream>>>(h, src, dst, dinv, agg, E, D);
  finalize_kernel<<<(unsigned)((nAgg + 255) / 256), 256, 0, stream>>>(agg, h, dinv, b, y, N, D);
}

extern "C" void kernel_launch(void* const* d_in, const int* in_sizes, int n_in,
                              void* d_out, int out_size, void* d_ws, size_t ws_size,
                              hipStream_t stream) {
  const float* x1  = (const float*)d_in[0];
  const int*   ei1 = (const int*)d_in[1];
  const float* x2  = (const float*)d_in[2];
  const int*   ei2 = (const int*)d_in[3];
  const float* W1  = (const float*)d_in[4];
  const float* b1  = (const float*)d_in[5];
  const float* W2  = (const float*)d_in[6];
  const float* b2  = (const float*)d_in[7];
  const float* W3  = (const float*)d_in[8];
  const float* b3  = (const float*)d_in[9];

  const int F0 = 128, D1 = 64, D2 = 64, D3 = 32;
  const int N1 = in_sizes[0] / F0;
  const int E1 = in_sizes[1] / 2;
  const int N2 = in_sizes[2] / F0;
  const int E2 = in_sizes[3] / 2;
  const int maxN = (N1 > N2) ? N1 : N2;

  const int* src1 = ei1;
  const int* dst1 = ei1 + E1;
  const int* src2 = ei2;
  const int* dst2 = ei2 + E2;

  // workspace carve-up (floats)
  float* ws = (float*)d_ws;
  size_t off = 0;
  float* dinv1 = ws + off; off += (size_t)N1;
  float* dinv2 = ws + off; off += (size_t)N2;
  float* hbuf  = ws + off; off += (size_t)maxN * 64;
  float* agg   = ws + off; off += (size_t)maxN * 64;
  float* y1s   = ws + off; off += (size_t)N1 * D1;
  float* y2s   = ws + off; off += (size_t)N1 * D2;
  float* y3s   = ws + off; off += (size_t)N1 * D3;
  float* y1t   = ws + off; off += (size_t)N2 * D1;
  float* y2t   = ws + off; off += (size_t)N2 * D2;
  float* y3t   = ws + off; off += (size_t)N2 * D3;
  float* U     = ws + off; off += (size_t)RESHAPE * 64;
  float* V     = ws + off; off += (size_t)RESHAPE * 64;
  (void)ws_size; (void)n_in;

  // --- symmetric-normalization degrees (deg = in-degree + self-loop) ---
  zero_f32_kernel<<<(N1 + 255) / 256, 256, 0, stream>>>(dinv1, N1);
  count_deg_kernel<<<(E1 + 255) / 256, 256, 0, stream>>>(dst1, dinv1, E1);
  deg_to_dinv_kernel<<<(N1 + 255) / 256, 256, 0, stream>>>(dinv1, N1);

  zero_f32_kernel<<<(N2 + 255) / 256, 256, 0, stream>>>(dinv2, N2);
  count_deg_kernel<<<(E2 + 255) / 256, 256, 0, stream>>>(dst2, dinv2, E2);
  deg_to_dinv_kernel<<<(N2 + 255) / 256, 256, 0, stream>>>(dinv2, N2);

  // --- 3-layer GCN, graph 1 ---
  run_gcn_layer(x1,  N1, F0, D1, src1, dst1, E1, W1, b1, dinv1, hbuf, agg, y1s, stream);
  run_gcn_layer(y1s, N1, D1, D2, src1, dst1, E1, W2, b2, dinv1, hbuf, agg, y2s, stream);
  run_gcn_layer(y2s, N1, D2, D3, src1, dst1, E1, W3, b3, dinv1, hbuf, agg, y3s, stream);

  // --- 3-layer GCN, graph 2 ---
  run_gcn_layer(x2,  N2, F0, D1, src2, dst2, E2, W1, b1, dinv2, hbuf, agg, y1t, stream);
  run_gcn_layer(y1t, N2, D1, D2, src2, dst2, E2, W2, b2, dinv2, hbuf, agg, y2t, stream);
  run_gcn_layer(y2t, N2, D2, D3, src2, dst2, E2, W3, b3, dinv2, hbuf, agg, y3t, stream);

  // --- resize(pad(xs) @ pad(xt)^T) == interp-rows(xs) @ interp-rows(xt)^T ---
  float* out = (float*)d_out;
  const float* Ys[3] = {y1s, y2s, y3s};
  const float* Yt[3] = {y1t, y2t, y3t};
  const int    Dl[3] = {D1, D2, D3};
  for (int l = 0; l < 3; ++l) {
    int D = Dl[l];
    interp_rows_kernel<<<RESHAPE, D, 0, stream>>>(Ys[l], N1, D, maxN, U);
    interp_rows_kernel<<<RESHAPE, D, 0, stream>>>(Yt[l], N2, D, maxN, V);
    gemm_nt_wmma<<<dim3(RESHAPE / 16, RESHAPE / 16), 32, 0, stream>>>(
        U, V, out + (size_t)l * RESHAPE * RESHAPE, RESHAPE, RESHAPE, D);
  }
  (void)out_size;
}